// SparseMultiHeadAttention_81896436400764
// MI455X (gfx1250) — compile-verified
//
#include <hip/hip_runtime.h>
#include <stdint.h>

typedef __attribute__((ext_vector_type(2))) float v2f;
typedef __attribute__((ext_vector_type(8))) float v8f;
typedef __attribute__((ext_vector_type(4))) unsigned int u32x4;
typedef __attribute__((ext_vector_type(8))) int i32x8;
typedef __attribute__((ext_vector_type(4))) int i32x4;

#define E_DIM 1024
#define M_LM  64
#define LQ    2048

// GEMM tiling
#define BM 128          // block rows
#define BN 64           // block cols
#define KC 32           // K chunk
#define LROW 34         // padded LDS row stride (32 data + 2 pad dwords)
#define A_ELEMS (BM * LROW)   // 4352
#define W_ELEMS (BN * LROW)   // 2176
#define BUF_ELEMS (A_ELEMS + W_ELEMS)

#if __has_builtin(__builtin_amdgcn_tensor_load_to_lds)
#define HAVE_TDM 1
#else
#define HAVE_TDM 0
#endif

static __device__ __forceinline__ v8f wmma_f32(v2f a, v2f b, v8f c) {
  // V_WMMA_F32_16X16X4_F32 : D = A(16x4) * B(4x16) + C(16x16), fp32
  return __builtin_amdgcn_wmma_f32_16x16x4_f32(
      false, a, false, b, (short)0, c, false, false);
}

#if HAVE_TDM
// Issue one TDM 2D tile load: rows x KC f32 elements, row stride `stride_elems`
// in memory, written to LDS at lds_byte_off with 2 pad dwords every 32 dwords
// (so LDS row stride = LROW floats).
static __device__ __forceinline__ void tdm_load_2d(
    const float* gptr, uint32_t lds_byte_off, uint32_t rows,
    uint32_t stride_elems)
{
  uint64_t ga = (uint64_t)(uintptr_t)gptr;
  u32x4 g0;
  g0.x = 1u;                                        // count=1, user mode
  g0.y = lds_byte_off;                              // lds_addr (bytes)
  g0.z = (uint32_t)(ga & 0xffffffffu);              // global_addr[31:0]
  g0.w = (uint32_t)((ga >> 32) & 0x01ffffffu)       // global_addr[56:32]
         | (2u << 30);                              // type = 2 ("image")
  i32x8 g1;
  uint32_t w0 = (2u << 16)                          // data_size = 4 bytes
              | (1u << 20)                          // pad_enable
              | (4u << 22)                          // pad_interval: 32 dwords
              | (1u << 25);                         // pad_amount: 2 dwords
  g1[0] = (int)w0;                                  // workgroup_mask = 0
  // tensor_dim0 = stride_elems (never OOB), bits 79:48
  g1[1] = (int)((stride_elems & 0xffffu) << 16);
  g1[2] = (int)(((stride_elems >> 16) & 0xffffu)    // tensor_dim0[31:16]
              | ((rows & 0xffffu) << 16));          // tensor_dim1[15:0]
  g1[3] = (int)(((rows >> 16) & 0xffffu)            // tensor_dim1[31:16]
              | ((uint32_t)KC << 16));              // tile_dim0 = KC
  g1[4] = (int)(rows & 0xffffu);                    // tile_dim1 = rows, tile_dim2=0
  g1[5] = (int)stride_elems;                        // tensor_dim0_stride[31:0]
  g1[6] = 0;                                        // stride[47:32], dim1_stride lo
  g1[7] = 0;
  i32x4 z4 = {0, 0, 0, 0};
  i32x8 z8 = {0, 0, 0, 0, 0, 0, 0, 0};
  // 6-arg form (clang-23 / therock-10.0 headers toolchain)
  __builtin_amdgcn_tensor_load_to_lds(g0, g1, z4, z4, z8, 0);
}
#endif

// C[R x N] = A[R x K] @ W[N x K]^T + bias[N]
// Block = 4 waves -> 128x64 tile; TDM double-buffers A/W K-chunks into LDS.
__global__ __launch_bounds__(128) void gemm_wt_bias(
    const float* __restrict__ A, const float* __restrict__ W,
    const float* __restrict__ bias, float* __restrict__ C,
    int R, int N, int K)
{
  __shared__ float smem[2 * BUF_ELEMS];   // ~51 KB

  const int lane  = threadIdx.x & 31;
  const int wave  = threadIdx.x >> 5;
  const int row16 = lane & 15;
  const int hi    = lane >> 4;

  const int tiles_n = N / BN;
  const int mt = blockIdx.x / tiles_n;
  const int nt = blockIdx.x % tiles_n;

  const float* Ab = A + (size_t)(mt * BM) * K;
  const float* Wb = W + (size_t)(nt * BN) * K;

#if HAVE_TDM
  const uint32_t lds0 = (uint32_t)(uintptr_t)smem;  // LDS byte offset of smem
#endif

  const int rbase = wave * 32;
  const int NC = K / KC;

  v8f acc[2][4] = {};

#if HAVE_TDM
  if (wave == 0) {
    tdm_load_2d(Ab, lds0, BM, (uint32_t)K);
    tdm_load_2d(Wb, lds0 + (uint32_t)A_ELEMS * 4u, BN, (uint32_t)K);
  }
#endif

  for (int c = 0; c < NC; ++c) {
    const int bofs = (c & 1) * BUF_ELEMS;
#if HAVE_TDM
    if (wave == 0) {
      if (c + 1 < NC) {
        const uint32_t boff = (uint32_t)(((c + 1) & 1) * BUF_ELEMS) * 4u;
        tdm_load_2d(Ab + (size_t)(c + 1) * KC, lds0 + boff, BM, (uint32_t)K);
        tdm_load_2d(Wb + (size_t)(c + 1) * KC,
                    lds0 + boff + (uint32_t)A_ELEMS * 4u, BN, (uint32_t)K);
        __builtin_amdgcn_s_wait_tensorcnt(2);   // chunk c done, c+1 in flight
      } else {
        __builtin_amdgcn_s_wait_tensorcnt(0);
      }
    }
    __syncthreads();
#else
    {
      // Fallback: cooperative copy of chunk c with padded rows.
      const float* Asrc = Ab + (size_t)c * KC;
      const float* Wsrc = Wb + (size_t)c * KC;
      float* dA = smem + bofs;
      float* dW = smem + bofs + A_ELEMS;
      for (int idx = threadIdx.x; idx < BM * KC; idx += 128) {
        const int r = idx >> 5, col = idx & 31;
        dA[r * LROW + col] = Asrc[(size_t)r * K + col];
      }
      for (int idx = threadIdx.x; idx < BN * KC; idx += 128) {
        const int r = idx >> 5, col = idx & 31;
        dW[r * LROW + col] = Wsrc[(size_t)r * K + col];
      }
      __syncthreads();
    }
#endif

    const float* pA = smem + bofs;
    const float* pW = smem + bofs + A_ELEMS;
#pragma unroll
    for (int kk = 0; kk < KC; kk += 4) {
      v2f a0 = *(const v2f*)(pA + (rbase + row16) * LROW + kk + 2 * hi);
      v2f a1 = *(const v2f*)(pA + (rbase + 16 + row16) * LROW + kk + 2 * hi);
#pragma unroll
      for (int j = 0; j < 4; ++j) {
        v2f b = *(const v2f*)(pW + (j * 16 + row16) * LROW + kk + 2 * hi);
        acc[0][j] = wmma_f32(a0, b, acc[0][j]);
        acc[1][j] = wmma_f32(a1, b, acc[1][j]);
      }
    }
    __syncthreads();   // protect buffer before next TDM overwrite
  }

#pragma unroll
  for (int sub = 0; sub < 2; ++sub) {
#pragma unroll
    for (int j = 0; j < 4; ++j) {
      const int col = nt * BN + j * 16 + row16;
      const float bn = bias[col];
      float* cp = C + (size_t)(mt * BM + rbase + sub * 16 + hi * 8) * N + col;
#pragma unroll
      for (int r = 0; r < 8; ++r)
        cp[(size_t)r * N] = acc[sub][j][r] + bn;
    }
  }
}

// OUT[R x 64] = softmax_rows( (X[R x 1024] @ Wl[64 x 1024]^T + bl) / 8 )
__global__ __launch_bounds__(256) void landmark_softmax(
    const float* __restrict__ X, const float* __restrict__ Wl,
    const float* __restrict__ bl, float* __restrict__ OUT)
{
  const int lane  = threadIdx.x & 31;
  const int wave  = threadIdx.x >> 5;
  const int row16 = lane & 15;
  const int hi    = lane >> 4;
  const int mt    = blockIdx.x * 8 + wave;

  const float* xrow = X + (size_t)(mt * 16 + row16) * E_DIM;

  v8f acc[4] = {};
  for (int k = 0; k < E_DIM; k += 8) {
    v2f a0 = *(const v2f*)(xrow + k + 2 * hi);
    v2f a1 = *(const v2f*)(xrow + k + 4 + 2 * hi);
    v2f b0[4], b1[4];
#pragma unroll
    for (int j = 0; j < 4; ++j) {
      const float* wr = Wl + (size_t)(j * 16 + row16) * E_DIM + k + 2 * hi;
      b0[j] = *(const v2f*)(wr);
      b1[j] = *(const v2f*)(wr + 4);
    }
#pragma unroll
    for (int j = 0; j < 4; ++j) acc[j] = wmma_f32(a0, b0[j], acc[j]);
#pragma unroll
    for (int j = 0; j < 4; ++j) acc[j] = wmma_f32(a1, b1[j], acc[j]);
  }

  float l[4][8];
#pragma unroll
  for (int j = 0; j < 4; ++j) {
    const float bj = bl[j * 16 + row16];
#pragma unroll
    for (int r = 0; r < 8; ++r)
      l[j][r] = (acc[j][r] + bj) * 0.125f;   // / sqrt(M=64)
  }

#pragma unroll
  for (int r = 0; r < 8; ++r) {
    float m = l[0][r];
#pragma unroll
    for (int j = 1; j < 4; ++j) m = fmaxf(m, l[j][r]);
    m = fmaxf(m, __shfl_xor(m, 1));
    m = fmaxf(m, __shfl_xor(m, 2));
    m = fmaxf(m, __shfl_xor(m, 4));
    m = fmaxf(m, __shfl_xor(m, 8));
    float s = 0.f;
#pragma unroll
    for (int j = 0; j < 4; ++j) { l[j][r] = __expf(l[j][r] - m); s += l[j][r]; }
    s += __shfl_xor(s, 1);
    s += __shfl_xor(s, 2);
    s += __shfl_xor(s, 4);
    s += __shfl_xor(s, 8);
    const float inv = 1.0f / s;
#pragma unroll
    for (int j = 0; j < 4; ++j) l[j][r] *= inv;
  }

#pragma unroll
  for (int j = 0; j < 4; ++j)
#pragma unroll
    for (int r = 0; r < 8; ++r)
      OUT[(size_t)(mt * 16 + hi * 8 + r) * M_LM + j * 16 + row16] = l[j][r];
}

// Fused landmark attention:  O1 = softmax_k( QL @ KL^T / 8 ) @ Vp   (per batch)
// S = <prob, prob>/8 in [0, 1/8] -> exp needs no max subtraction: single pass.
__global__ __launch_bounds__(128) void nystrom_attn(
    const float* __restrict__ QL, const float* __restrict__ KL,
    const float* __restrict__ Vp, float* __restrict__ O, int Lk)
{
  __shared__ float pbuf[4][16 * 16];

  const int lane  = threadIdx.x & 31;
  const int wave  = threadIdx.x >> 5;
  const int row16 = lane & 15;
  const int hi    = lane >> 4;

  const int qrow0 = blockIdx.x * 16;
  const int batch = qrow0 / LQ;
  const int cbase = blockIdx.y * 64 + wave * 16;

  // QL A-fragments (16 rows x K=64), reused for every key tile
  v2f aq[16];
  const float* qlrow = QL + (size_t)(qrow0 + row16) * M_LM;
#pragma unroll
  for (int kc = 0; kc < 16; ++kc)
    aq[kc] = *(const v2f*)(qlrow + kc * 4 + 2 * hi);

  const float* KLb = KL + (size_t)batch * Lk * M_LM;
  const float* Vb  = Vp + (size_t)batch * Lk * E_DIM;
  float* pw = &pbuf[wave][0];

  v8f o = {};
  float rsum[8] = {0.f, 0.f, 0.f, 0.f, 0.f, 0.f, 0.f, 0.f};

  for (int k0 = 0; k0 < Lk; k0 += 16) {
    // Hoist all operand loads for this tile so waits amortize over chains.
    const float* klrow = KLb + (size_t)(k0 + row16) * M_LM;
    v2f bk[16];
#pragma unroll
    for (int kc = 0; kc < 16; ++kc)
      bk[kc] = *(const v2f*)(klrow + kc * 4 + 2 * hi);

    v2f bv4[4];
#pragma unroll
    for (int ks = 0; ks < 4; ++ks) {
      const float* vr =
          Vb + (size_t)(k0 + ks * 4 + 2 * hi) * E_DIM + cbase + row16;
      bv4[ks].x = vr[0];
      bv4[ks].y = vr[E_DIM];
    }

    // S(16x16) = QL_tile @ KL_tile^T, contraction over 64 landmarks
    v8f s = {};
#pragma unroll
    for (int kc = 0; kc < 16; ++kc)
      s = wmma_f32(aq[kc], bk[kc], s);

    // P = exp(S/8); stage to LDS (acc layout -> A layout); accumulate row sums
#pragma unroll
    for (int r = 0; r < 8; ++r) {
      const float e = __expf(s[r] * 0.125f);   // / sqrt(HD=64)
      rsum[r] += e;
      pw[(r + 8 * hi) * 16 + row16] = e;
    }

    // O += P(16x16) @ Vp_tile(16 x 16 cols)
#pragma unroll
    for (int ks = 0; ks < 4; ++ks) {
      v2f a = *(const v2f*)(pw + row16 * 16 + ks * 4 + 2 * hi);
      o = wmma_f32(a, bv4[ks], o);
    }
  }

#pragma unroll
  for (int r = 0; r < 8; ++r) {
    float s = rsum[r];
    s += __shfl_xor(s, 1);
    s += __shfl_xor(s, 2);
    s += __shfl_xor(s, 4);
    s += __shfl_xor(s, 8);
    const float inv = 1.0f / s;
    O[(size_t)(qrow0 + hi * 8 + r) * E_DIM + cbase + row16] = o[r] * inv;
  }
}

extern "C" void kernel_launch(void* const* d_in, const int* in_sizes, int n_in,
                              void* d_out, int out_size, void* d_ws, size_t ws_size,
                              hipStream_t stream) {
  const float* query = (const float*)d_in[0];
  const float* key   = (const float*)d_in[1];
  const float* value = (const float*)d_in[2];
  const float* Wv    = (const float*)d_in[3];
  const float* bv    = (const float*)d_in[4];
  const float* Wl    = (const float*)d_in[5];
  const float* bl    = (const float*)d_in[6];
  const float* Wo    = (const float*)d_in[7];
  const float* bo    = (const float*)d_in[8];
  float* out = (float*)d_out;

  const int R  = 4096;   // B * L
  const int Lk = LQ;

  float* ws  = (float*)d_ws;
  float* Vp  = ws;                              // [4096 x 1024]
  float* QL  = Vp + (size_t)R * E_DIM;          // [4096 x 64]
  float* KLm = QL + (size_t)R * M_LM;           // [4096 x 64]
  float* O1  = KLm + (size_t)R * M_LM;          // [4096 x 1024]

  const int gemm_blocks = (R / BM) * (E_DIM / BN);   // 32 * 16 = 512
  const int lm_blocks   = (R / 16) / 8;              // 32

  // 1) Vp = value @ Wv^T + bv
  gemm_wt_bias<<<gemm_blocks, 128, 0, stream>>>(value, Wv, bv, Vp, R, E_DIM, E_DIM);
  // 2,3) landmark projections + softmax
  landmark_softmax<<<lm_blocks, 256, 0, stream>>>(query, Wl, bl, QL);
  landmark_softmax<<<lm_blocks, 256, 0, stream>>>(key,   Wl, bl, KLm);
  // 4) fused landmark attention
  nystrom_attn<<<dim3(R / 16, E_DIM / 64), 128, 0, stream>>>(QL, KLm, Vp, O1, Lk);
  // 5) out = O1 @ Wo^T + bo
  gemm_wt_bias<<<gemm_blocks, 128, 0, stream>>>(O1, Wo, bo, out, R, E_DIM, E_DIM);
}